// LocalAttentionMaskProvider2d_37933151158508
// MI455X (gfx1250) — compile-verified
//
#include <hip/hip_runtime.h>
#include <hip/hip_bf16.h>

// Reference: local attention mask for H=64, W=160, KH=7, KW=15.
// Output: (10240 x 10240) float mask, 1.0 = blocked, 0.0 = attend.
// Pure store-bandwidth-bound kernel: 419.4 MB of output, 0 input bytes,
// 0 FLOPs -> optimal path is coalesced b128 non-temporal stores (output is
// 2.2x the 192MB L2, so NT avoids thrashing). WMMA/TDM not applicable.

namespace {
constexpr int H  = 64;
constexpr int W  = 160;
constexpr int N  = H * W;          // 10240
constexpr int KH = 7;
constexpr int KW = 15;
constexpr int RH = KH / 2;         // 3
constexpr int RW = KW / 2;         // 7
constexpr int BLK        = 256;    // 8 wave32 per block
constexpr int GROUPS     = N / 4;  // 2560 float4 groups per row
constexpr int PER_THREAD = GROUPS / BLK;  // 10
}

typedef float v4f __attribute__((ext_vector_type(4)));

__global__ __launch_bounds__(BLK)
void LocalAttentionMask2d_kernel(float* __restrict__ out) {
    // One block per output row i. Row-invariant values stay scalar (SGPR).
    const int i   = (int)blockIdx.x;          // 0 .. N-1
    const int h_i = i / W;                    // scalar magic-number div
    const int w_i = i - h_i * W;
    float* __restrict__ row = out + (size_t)i * (size_t)N;

    const int t = (int)threadIdx.x;

    #pragma unroll
    for (int it = 0; it < PER_THREAD; ++it) {
        const int g  = t + it * BLK;          // float4 group index in row
        const int j  = g * 4;                 // column of first element
        // W % 4 == 0 and j % 4 == 0  =>  all 4 columns share h_j.
        const int h_j = j / W;
        const int w_j = j - h_j * W;

        const bool ok_h = (unsigned)(h_j - h_i + RH) < (unsigned)KH;
        const int  base = w_j - w_i + RW;     // ok_w iff 0 <= base+k < KW

        v4f v;
        v.x = (ok_h && (unsigned)(base + 0) < (unsigned)KW) ? 0.0f : 1.0f;
        v.y = (ok_h && (unsigned)(base + 1) < (unsigned)KW) ? 0.0f : 1.0f;
        v.z = (ok_h && (unsigned)(base + 2) < (unsigned)KW) ? 0.0f : 1.0f;
        v.w = (ok_h && (unsigned)(base + 3) < (unsigned)KW) ? 0.0f : 1.0f;

        // Non-temporal 128-bit store: global_store_b128 with NT hint.
        __builtin_nontemporal_store(v, (v4f*)(row + j));
    }
}

extern "C" void kernel_launch(void* const* d_in, const int* in_sizes, int n_in,
                              void* d_out, int out_size, void* d_ws, size_t ws_size,
                              hipStream_t stream) {
    (void)d_in; (void)in_sizes; (void)n_in; (void)d_ws; (void)ws_size; (void)out_size;
    float* out = (float*)d_out;
    // 10240 blocks x 256 threads; each block emits one full 40KB output row.
    LocalAttentionMask2d_kernel<<<dim3(N), dim3(BLK), 0, stream>>>(out);
}